// InstanceHead_65274912964728
// MI455X (gfx1250) — compile-verified
//
#include <hip/hip_runtime.h>
#include <math.h>

typedef float v2f __attribute__((ext_vector_type(2)));
typedef float v8f __attribute__((ext_vector_type(8)));

#define N_  65536
#define P_  128
#define D_  16
#define C_  512
#define EPS_ 1e-5f

// workspace layout (float offsets)
#define WS_SEG   0
#define WS_CNT   (WS_SEG + C_*D_)          // 8192
#define WS_DESC  (WS_CNT + C_)             // 8704
#define WS_H     (WS_DESC + C_*D_)         // 16896
#define WS_HV    (WS_H  + N_*D_)
#define WS_HN2   (WS_HV + N_*D_)
#define WS_CENT  (WS_HN2 + N_)
#define WS_CC    (WS_CENT + P_*D_)
#define WS_CN2   (WS_CC + P_*D_)
#define WS_CB    (WS_CN2 + P_)             // ints in float slots

__global__ void zero_kernel(float* ws) {
    int i = blockIdx.x * 256 + threadIdx.x;        // 34*256 == 8704 exactly
    ws[i] = 0.0f;
}

// LDS-privatized segment sum: 64 blocks x 256 threads, 1024 rows per block
__global__ void seg_kernel(const float* __restrict__ feats,
                           const int* __restrict__ cmap, float* ws) {
    __shared__ float sAcc[C_ * D_];
    __shared__ float sCnt[C_];
    for (int i = threadIdx.x; i < C_ * D_; i += 256) sAcc[i] = 0.0f;
    for (int i = threadIdx.x; i < C_;      i += 256) sCnt[i] = 0.0f;
    __syncthreads();
    int base = blockIdx.x * 1024;
    for (int it = 0; it < 4; ++it) {
        int n = base + it * 256 + threadIdx.x;
        int c = cmap[n];
        const float* f = feats + (size_t)n * D_;
        #pragma unroll
        for (int d = 0; d < D_; ++d) atomicAdd(&sAcc[c * D_ + d], f[d]);
        atomicAdd(&sCnt[c], 1.0f);
    }
    __syncthreads();
    for (int i = threadIdx.x; i < C_ * D_; i += 256) atomicAdd(&ws[WS_SEG + i], sAcc[i]);
    for (int i = threadIdx.x; i < C_;      i += 256) atomicAdd(&ws[WS_CNT + i], sCnt[i]);
}

__global__ void desc_kernel(float* ws) {
    int i = blockIdx.x * 256 + threadIdx.x;        // 32*256 == 8192
    float cnt = ws[WS_CNT + (i >> 4)];
    ws[WS_DESC + i] = ws[WS_SEG + i] / fmaxf(cnt, 1.0f);
}

// h = relu(bn1(concat(feats, desc[cmap]) @ W1 + b1)); also hv = h @ Wv, hn2 = |h|^2
__global__ void __launch_bounds__(256) h_kernel(
    const float* __restrict__ feats, const int* __restrict__ cmap,
    const float* __restrict__ W1, const float* __restrict__ b1,
    const float* __restrict__ g1, const float* __restrict__ be1,
    const float* __restrict__ m1, const float* __restrict__ v1,
    const float* __restrict__ W2a, float* ws) {
    __shared__ float sW1[32 * 16];
    __shared__ float sWv[16 * 16];
    __shared__ float sS1[16], sT1[16], sB1[16];
    int t = threadIdx.x;
    for (int i = t; i < 512; i += 256) sW1[i] = W1[i];
    if (t < 256) sWv[t] = W2a[t];                    // W2a rows [0:16) = Wv
    if (t < 16) {
        float s = g1[t] * rsqrtf(v1[t] + EPS_);
        sS1[t] = s; sT1[t] = be1[t] - m1[t] * s; sB1[t] = b1[t];
    }
    __syncthreads();

    int n = blockIdx.x * 256 + t;
    const float* f   = feats + (size_t)n * 16;
    const float* dsc = ws + WS_DESC + (size_t)cmap[n] * 16;
    float x[32];
    #pragma unroll
    for (int k = 0; k < 16; ++k) x[k] = f[k];
    #pragma unroll
    for (int k = 0; k < 16; ++k) x[16 + k] = dsc[k];

    float h[16]; float hn2 = 0.0f;
    #pragma unroll
    for (int d = 0; d < 16; ++d) {
        float acc = sB1[d];
        #pragma unroll
        for (int k = 0; k < 32; ++k) acc += x[k] * sW1[k * 16 + d];
        acc = fmaxf(acc * sS1[d] + sT1[d], 0.0f);
        h[d] = acc; hn2 += acc * acc;
    }
    float* hrow  = ws + WS_H  + (size_t)n * 16;
    float* hvrow = ws + WS_HV + (size_t)n * 16;
    #pragma unroll
    for (int d = 0; d < 16; ++d) hrow[d] = h[d];
    #pragma unroll
    for (int dp = 0; dp < 16; ++dp) {
        float acc = 0.0f;
        #pragma unroll
        for (int d = 0; d < 16; ++d) acc += h[d] * sWv[d * 16 + dp];
        hvrow[dp] = acc;
    }
    ws[WS_HN2 + n] = hn2;
}

// cent = h[peaks]*conf; cc = cent @ Wc; cn2 = |cent|^2; cb = batch_idx[peaks]
__global__ void cent_kernel(const int* __restrict__ batch_idx,
                            const int* __restrict__ peaks,
                            const float* __restrict__ conf,
                            const float* __restrict__ W2a, float* ws) {
    int p = threadIdx.x;                             // 128 threads
    int idx = peaks[p];
    float cf = conf[p];
    const float* hrow = ws + WS_H + (size_t)idx * 16;
    float c[16]; float cn2 = 0.0f;
    #pragma unroll
    for (int d = 0; d < 16; ++d) { c[d] = hrow[d] * cf; cn2 += c[d] * c[d]; }
    float* crow  = ws + WS_CENT + p * 16;
    float* ccrow = ws + WS_CC   + p * 16;
    #pragma unroll
    for (int d = 0; d < 16; ++d) crow[d] = c[d];
    #pragma unroll
    for (int dp = 0; dp < 16; ++dp) {
        float acc = 0.0f;
        #pragma unroll
        for (int d = 0; d < 16; ++d) acc += c[d] * W2a[(16 + d) * 16 + dp];
        ccrow[dp] = acc;
    }
    ws[WS_CN2 + p] = cn2;
    ((int*)ws)[WS_CB + p] = batch_idx[idx];
}

// Main: one block per 16-row tile; wave w handles column tile w (P=128 -> 8 tiles).
// Gram matrix G = h_tile x cent_tile^T via 4 chained V_WMMA_F32_16X16X4_F32.
__global__ void __launch_bounds__(256) score_kernel(
    const int* __restrict__ batch_idx, const float* __restrict__ W2a,
    const float* __restrict__ g2, const float* __restrict__ be2,
    const float* __restrict__ m2, const float* __restrict__ v2,
    const float* __restrict__ W2b, const float* __restrict__ b2,
    const float* __restrict__ ws, float* __restrict__ out) {
    __shared__ float sH[16][17];
    __shared__ float sHv[16][17];
    __shared__ float sHn2[16];
    __shared__ int   sBidx[16];
    __shared__ float sC[128][17];
    __shared__ float sCc[128][17];
    __shared__ float sCn2[128];
    __shared__ int   sCb[128];
    __shared__ float sWd[16], sS2[16], sT2[16], sW2b[16];
    __shared__ float sB2;

    int t = threadIdx.x;
    int rowBase = blockIdx.x * 16;
    {
        int r = t >> 4, c = t & 15;                  // 256 == 16x16
        sH[r][c]  = ws[WS_H  + (size_t)(rowBase + r) * 16 + c];
        sHv[r][c] = ws[WS_HV + (size_t)(rowBase + r) * 16 + c];
    }
    for (int i = t; i < 2048; i += 256) {
        int r = i >> 4, c = i & 15;
        sC[r][c]  = ws[WS_CENT + i];
        sCc[r][c] = ws[WS_CC + i];
    }
    if (t < 16) {
        sHn2[t]  = ws[WS_HN2 + rowBase + t];
        sBidx[t] = batch_idx[rowBase + t];
        sWd[t]   = W2a[512 + t];                     // W2a row 32 = wd
        float s  = g2[t] * rsqrtf(v2[t] + EPS_);
        sS2[t] = s; sT2[t] = be2[t] - m2[t] * s;
        sW2b[t] = W2b[t];
    }
    if (t < 128) { sCn2[t] = ws[WS_CN2 + t]; sCb[t] = ((const int*)ws)[WS_CB + t]; }
    if (t == 0) sB2 = b2[0];
    __syncthreads();

    int lane = t & 31;
    int wave = t >> 5;                               // column tile 0..7
    int hi   = lane >> 4;                            // 0 -> K 0,1 ; 1 -> K 2,3
    int lid  = lane & 15;
    int pBase = wave * 16;

    v8f acc = {};
    #pragma unroll
    for (int k0 = 0; k0 < 16; k0 += 4) {
        v2f a, b;
        a.x = sH[lid][k0 + 2 * hi];
        a.y = sH[lid][k0 + 2 * hi + 1];
        b.x = sC[pBase + lid][k0 + 2 * hi];
        b.y = sC[pBase + lid][k0 + 2 * hi + 1];
        acc = __builtin_amdgcn_wmma_f32_16x16x4_f32(
            false, a, false, b, (short)0, acc, false, false);
    }

    int p = pBase + lid;
    float cn2 = sCn2[p];
    int   cb  = sCb[p];
    #pragma unroll
    for (int j = 0; j < 8; ++j) {
        int m = j + 8 * hi;                          // C/D layout: lanes0-15 M=j, lanes16-31 M=j+8
        float g  = acc[j];
        float d2 = sHn2[m] + cn2 - 2.0f * g;
        float dist = fmaxf(sqrtf(fmaxf(d2, 0.0f)), 1.0f);
        float score = 0.0f;
        #pragma unroll
        for (int d = 0; d < 16; ++d) {
            float z = sHv[m][d] + sCc[p][d] + dist * sWd[d];
            z = fmaxf(z * sS2[d] + sT2[d], 0.0f);
            score += z * sW2b[d];
        }
        score += sB2;
        int n = rowBase + m;
        out[(size_t)n * P_ + p] = (sBidx[m] == cb) ? score : -__builtin_huge_valf();
    }
}

extern "C" void kernel_launch(void* const* d_in, const int* in_sizes, int n_in,
                              void* d_out, int out_size, void* d_ws, size_t ws_size,
                              hipStream_t stream) {
    const float* feats = (const float*)d_in[0];
    const int*   bidx  = (const int*)d_in[1];
    const int*   cmap  = (const int*)d_in[2];
    const int*   peaks = (const int*)d_in[3];
    const float* conf  = (const float*)d_in[4];
    const float* W1    = (const float*)d_in[5];
    const float* b1    = (const float*)d_in[6];
    const float* g1    = (const float*)d_in[7];
    const float* be1   = (const float*)d_in[8];
    const float* m1    = (const float*)d_in[9];
    const float* v1    = (const float*)d_in[10];
    const float* W2a   = (const float*)d_in[11];
    const float* g2    = (const float*)d_in[12];
    const float* be2   = (const float*)d_in[13];
    const float* m2    = (const float*)d_in[14];
    const float* v2    = (const float*)d_in[15];
    const float* W2b   = (const float*)d_in[16];
    const float* b2    = (const float*)d_in[17];
    float* ws  = (float*)d_ws;
    float* out = (float*)d_out;

    zero_kernel <<<34,   256, 0, stream>>>(ws);
    seg_kernel  <<<64,   256, 0, stream>>>(feats, cmap, ws);
    desc_kernel <<<32,   256, 0, stream>>>(ws);
    h_kernel    <<<256,  256, 0, stream>>>(feats, cmap, W1, b1, g1, be1, m1, v1, W2a, ws);
    cent_kernel <<<1,    128, 0, stream>>>(bidx, peaks, conf, W2a, ws);
    score_kernel<<<N_/16,256, 0, stream>>>(bidx, W2a, g2, be2, m2, v2, W2b, b2, ws, out);
}